// Sylo_20014547599830
// MI455X (gfx1250) — compile-verified
//
#include <hip/hip_runtime.h>

typedef __attribute__((ext_vector_type(2))) float v2f;
typedef __attribute__((ext_vector_type(8))) float v8f;

#define NN 4
#define CI 4
#define CO 8
#define HH 512
#define WW 512
#define RK 4

// ---------------------------------------------------------------------------
// Kernel 1: S_row[n,o,h] = sum_i sum_r L[o,i,h,r] * (sum_w x[n,i,h,w]*L[o,i,w,r])
// Block = 256 threads (8 waves). Each block: one n, 64 h rows (8 per wave),
// one 128-wide w chunk. L slice for the chunk is staged in LDS (64 KB) so the
// hot inner loop never touches L2 for L. Partial sums combined with atomics.
// ---------------------------------------------------------------------------
__global__ __launch_bounds__(256) void sylo_rowsum(const float* __restrict__ x,
                                                   const float* __restrict__ L,
                                                   float* __restrict__ S_row) {
  __shared__ float4 lds[CI * CO * 128];   // 64 KB: L[o,i, wchunk, r] as float4
  const int tid  = threadIdx.x;
  const int lane = tid & 31;
  const int wave = tid >> 5;

  int b = blockIdx.x;
  const int wc = b % (WW / 128); b /= (WW / 128);
  const int hb = b % (HH / 64);  b /= (HH / 64);
  const int n  = b;

  // cooperative load of the L slice: all (i,o), 128 w's, all r
#pragma unroll
  for (int k = 0; k < 16; ++k) {
    const int f  = tid + k * 256;       // 0..4095 float4s
    const int wl = f & 127;
    const int oi = f >> 7;              // 0..31
    const int o  = oi >> 2, i = oi & 3;
    lds[(i * CO + o) * 128 + wl] =
        ((const float4*)L)[(o * CI + i) * HH + wc * 128 + wl];
  }
  __syncthreads();

  for (int hr = 0; hr < 8; ++hr) {
    const int h = hb * 64 + wave * 8 + hr;
    float s[CO];
#pragma unroll
    for (int o = 0; o < CO; ++o) s[o] = 0.f;

#pragma unroll
    for (int i = 0; i < CI; ++i) {
      float acc[CO][4];
#pragma unroll
      for (int o = 0; o < CO; ++o)
#pragma unroll
        for (int r = 0; r < 4; ++r) acc[o][r] = 0.f;

      const float* xrow = x + ((size_t)(n * CI + i) * HH + h) * WW + wc * 128;
#pragma unroll
      for (int wi = 0; wi < 4; ++wi) {
        const float xv = xrow[wi * 32 + lane];   // coalesced 128B per wave
#pragma unroll
        for (int o = 0; o < CO; ++o) {
          const float4 l4 = lds[(i * CO + o) * 128 + wi * 32 + lane];
          acc[o][0] += xv * l4.x;
          acc[o][1] += xv * l4.y;
          acc[o][2] += xv * l4.z;
          acc[o][3] += xv * l4.w;
        }
      }
#pragma unroll
      for (int o = 0; o < CO; ++o) {
        const float4 lh = ((const float4*)L)[(o * CI + i) * HH + h]; // broadcast
        s[o] += lh.x * acc[o][0] + lh.y * acc[o][1] +
                lh.z * acc[o][2] + lh.w * acc[o][3];
      }
    }
    // wave32 butterfly reduction over the lane-partitioned w axis
#pragma unroll
    for (int o = 0; o < CO; ++o) {
      float v = s[o];
      for (int off = 16; off > 0; off >>= 1) v += __shfl_xor(v, off, 32);
      if (lane == 0) atomicAdd(&S_row[(n * CO + o) * HH + h], v);
    }
  }
}

// ---------------------------------------------------------------------------
// Kernel 2: S_col[n,o,w] = sum_i sum_r L[o,i,w,r] * (sum_h x[n,i,h,w]*L[o,i,h,r])
// Lane <-> w (coalesced x reads down columns of rows), loop over an h chunk.
// L[o,i,h,:] is lane-invariant per h -> broadcast loads. Grid partitions
// (n, i, h-chunk, 32-wide w chunk); partials combined with float atomics.
// ---------------------------------------------------------------------------
__global__ __launch_bounds__(256) void sylo_colsum(const float* __restrict__ x,
                                                   const float* __restrict__ L,
                                                   float* __restrict__ S_col) {
  const int tid  = threadIdx.x;
  const int lane = tid & 31;
  const int wave = tid >> 5;

  int g = blockIdx.x * 8 + wave;                 // 0 .. NN*CI*4*(WW/32)-1
  const int wch = g % (WW / 32); g /= (WW / 32);
  const int hs  = g % 4;         g /= 4;
  const int i   = g % CI;        g /= CI;
  const int n   = g;

  const int w = wch * 32 + lane;

  float acc[CO][4];
#pragma unroll
  for (int o = 0; o < CO; ++o)
#pragma unroll
    for (int r = 0; r < 4; ++r) acc[o][r] = 0.f;

  const float* xp = x + ((size_t)(n * CI + i) * HH + hs * 128) * WW + w;
  for (int hh = 0; hh < 128; ++hh) {
    const float xv = xp[(size_t)hh * WW];        // coalesced across lanes
    const int h = hs * 128 + hh;
#pragma unroll
    for (int o = 0; o < CO; ++o) {
      const float4 lh = ((const float4*)L)[(o * CI + i) * HH + h]; // broadcast
      acc[o][0] += xv * lh.x;
      acc[o][1] += xv * lh.y;
      acc[o][2] += xv * lh.z;
      acc[o][3] += xv * lh.w;
    }
  }
#pragma unroll
  for (int o = 0; o < CO; ++o) {
    const float4 lw = ((const float4*)L)[(o * CI + i) * HH + w];   // per-lane
    const float v = lw.x * acc[o][0] + lw.y * acc[o][1] +
                    lw.z * acc[o][2] + lw.w * acc[o][3];
    atomicAdd(&S_col[(n * CO + o) * WW + w], v);
  }
}

// ---------------------------------------------------------------------------
// Kernel 3 (fused output, WMMA): per 16x16 tile and per (i,o):
//   T_tile = Lh(16x4) @ Lw(4x16)^T  via  V_WMMA_F32_16X16X4_F32
//   p     += x_tile[i] .* T_tile    (elementwise, in C/D register layout)
//   out    = S_row + S_col + bias - p
// Block = 256 (8 waves); wave w handles output channel o = w; the x tile for
// all C_IN is staged once in LDS and shared by the 8 waves.
// ---------------------------------------------------------------------------
__global__ __launch_bounds__(256) void sylo_main(const float* __restrict__ x,
                                                 const float* __restrict__ L,
                                                 const float* __restrict__ bias,
                                                 const float* __restrict__ S_row,
                                                 const float* __restrict__ S_col,
                                                 float* __restrict__ out) {
  __shared__ float xs[CI][16 * 16];
  const int tid  = threadIdx.x;
  const int lane = tid & 31;
  const int o    = tid >> 5;                     // wave id == output channel

  int b = blockIdx.x;
  const int tw = b % (WW / 16); b /= (WW / 16);
  const int th = b % (HH / 16); b /= (HH / 16);
  const int n  = b;

  {
    const int col = tid & 15;
    const int row = (tid >> 4) & 15;
#pragma unroll
    for (int i = 0; i < CI; ++i)
      xs[i][row * 16 + col] =
          x[((size_t)(n * CI + i) * HH + th * 16 + row) * WW + tw * 16 + col];
  }
  __syncthreads();

  const int m  = lane & 15;   // A: row M;  B/C/D: column N
  const int hi = lane >> 4;   // half-wave select
  const int kb = hi * 2;      // K base: {0,1} or {2,3}

  v8f p = {0.f, 0.f, 0.f, 0.f, 0.f, 0.f, 0.f, 0.f};
#pragma unroll
  for (int i = 0; i < CI; ++i) {
    const float* Lp = L + (size_t)((o * CI + i) * HH) * RK;
    v2f a, bm;
    a.x  = Lp[(th * 16 + m) * RK + kb];       // A[m,k]   = L[o,i,h0+m,k]
    a.y  = Lp[(th * 16 + m) * RK + kb + 1];
    bm.x = Lp[(tw * 16 + m) * RK + kb];       // B[k,n]   = L[o,i,w0+n,k]
    bm.y = Lp[(tw * 16 + m) * RK + kb + 1];
    v8f t = {};
    t = __builtin_amdgcn_wmma_f32_16x16x4_f32(
        /*neg_a=*/false, a, /*neg_b=*/false, bm,
        /*c_mod=*/(short)0, t, /*reuse_a=*/false, /*reuse_b=*/false);
#pragma unroll
    for (int v = 0; v < 8; ++v)
      p[v] += xs[i][(v + 8 * hi) * 16 + m] * t[v];   // x .* T, C/D layout
  }

  const float bo = bias[o];
  const float sc = S_col[(n * CO + o) * WW + tw * 16 + m];
  const float* srp = S_row + (n * CO + o) * HH + th * 16 + 8 * hi;
  float* op = out + ((size_t)(n * CO + o) * HH + th * 16 + 8 * hi) * WW +
              tw * 16 + m;
#pragma unroll
  for (int v = 0; v < 8; ++v)
    op[(size_t)v * WW] = srp[v] + sc + bo - p[v];
}

// ---------------------------------------------------------------------------
extern "C" void kernel_launch(void* const* d_in, const int* in_sizes, int n_in,
                              void* d_out, int out_size, void* d_ws, size_t ws_size,
                              hipStream_t stream) {
  (void)in_sizes; (void)n_in; (void)out_size; (void)ws_size;
  const float* x    = (const float*)d_in[0];
  const float* L    = (const float*)d_in[1];
  const float* bias = (const float*)d_in[2];
  float* out   = (float*)d_out;
  float* S_row = (float*)d_ws;                   // NN*CO*HH floats
  float* S_col = S_row + NN * CO * HH;           // NN*CO*WW floats

  hipMemsetAsync(d_ws, 0, (size_t)2 * NN * CO * HH * sizeof(float), stream);

  sylo_rowsum<<<NN * (HH / 64) * (WW / 128), 256, 0, stream>>>(x, L, S_row);
  sylo_colsum<<<(NN * CI * 4 * (WW / 32)) / 8, 256, 0, stream>>>(x, L, S_col);
  sylo_main<<<NN * (HH / 16) * (WW / 16), 256, 0, stream>>>(x, L, bias,
                                                            S_row, S_col, out);
}